// ModifiedAFFWithSoftmax_5291399709474
// MI455X (gfx1250) — compile-verified
//
#include <hip/hip_runtime.h>
#include <hip/hip_bf16.h>

typedef __attribute__((ext_vector_type(2))) float v2f;
typedef __attribute__((ext_vector_type(8))) float v8f;

#define BN_EPS   1e-5f
#define SPATIAL  (512 * 512)        // H*W = 262144 (power of two)
#define NPIX     (32 * SPATIAL)     // B*H*W = 8388608
#define NBLK     (NPIX / 32)        // 32-pixel blocks per wave iteration

// D = A(16x4) * B(4x16) + C, fp32, wave32.
// A layout: VGPR0 = K0 (lanes 0-15) / K2 (lanes 16-31), VGPR1 = K1 / K3.
// B layout: VGPR0 = row K0 (lanes 0-15) / row K2 (lanes 16-31), VGPR1 = K1 / K3.
// C/D layout: VGPR m = row m (lanes 0-15) / row 8+m (lanes 16-31), col = lane&15.
__device__ __forceinline__ v8f wmma4(v2f a, v2f b, v8f c) {
    return __builtin_amdgcn_wmma_f32_16x16x4_f32(
        /*neg_a=*/false, a, /*neg_b=*/false, b,
        /*c_mod=*/(short)0, c, /*reuse_a=*/false, /*reuse_b=*/false);
}

__global__ __launch_bounds__(256) void aff_softmax_kernel(
    const float* __restrict__ x,
    const float* __restrict__ w1, const float* __restrict__ b1,
    const float* __restrict__ g1, const float* __restrict__ be1,
    const float* __restrict__ m1, const float* __restrict__ v1,
    const float* __restrict__ w2, const float* __restrict__ b2,
    const float* __restrict__ g2, const float* __restrict__ be2,
    const float* __restrict__ m2, const float* __restrict__ v2,
    float* __restrict__ out)
{
    const int lane = threadIdx.x & 31;
    const int m    = lane & 15;         // A-matrix row / B-matrix column
    const bool hi  = lane >= 16;

    // ---------------- per-lane constant setup (amortized over grid-stride loop) ----
    // BN folded into conv weights: We[o][c] = w[o][c] * g[o]/sqrt(v[o]+eps)
    // BN+conv bias folded into WMMA C accumulator init.
    v2f A1a = {0.f, 0.f};   // conv1, K = channels 0..3
    v2f A1b = {0.f, 0.f};   // conv1, K = channels 4..5 (6,7 zero)
    v2f A2  = {0.f, 0.f};   // conv2, K = channels 0..2 (3 zero)
    if (m < 3) {
        float inv = g1[m] * rsqrtf(v1[m] + BN_EPS);
        A1a.x = w1[m * 6 + (hi ? 2 : 0)] * inv;
        A1a.y = w1[m * 6 + (hi ? 3 : 1)] * inv;
        if (!hi) { A1b.x = w1[m * 6 + 4] * inv; A1b.y = w1[m * 6 + 5] * inv; }
    }
    if (m < 6) {
        float inv = g2[m] * rsqrtf(v2[m] + BN_EPS);
        A2.x = w2[m * 3 + (hi ? 2 : 0)] * inv;
        if (!hi) A2.y = w2[m * 3 + 1] * inv;
    }
    v8f C1 = {0.f,0.f,0.f,0.f,0.f,0.f,0.f,0.f};
    #pragma unroll
    for (int o = 0; o < 3; ++o) {
        float inv = g1[o] * rsqrtf(v1[o] + BN_EPS);
        C1[o] = b1[o] * inv + be1[o] - m1[o] * inv;
    }
    v8f C2 = {0.f,0.f,0.f,0.f,0.f,0.f,0.f,0.f};
    #pragma unroll
    for (int k = 0; k < 6; ++k) {
        float inv = g2[k] * rsqrtf(v2[k] + BN_EPS);
        C2[k] = b2[k] * inv + be2[k] - m2[k] * inv;
    }

    const int c_b0 = hi ? 2 : 0;        // channel feeding B VGPR0 (rows K0/K2)
    const int c_b1 = hi ? 3 : 1;        // channel feeding B VGPR1 (rows K1/K3)

    // wave-uniform grid-stride loop over 32-pixel blocks (EXEC stays all-ones)
    const int wave  = blockIdx.x * (blockDim.x >> 5) + (threadIdx.x >> 5);
    const int nwave = gridDim.x * (blockDim.x >> 5);

    for (int blk = wave; blk < NBLK; blk += nwave) {
        const int base = blk << 5;                    // first pixel of block
        const int b    = base >> 18;                  // SPATIAL == 1<<18
        const int soff = base & (SPATIAL - 1);        // blocks never straddle batches
        const float* xb = x + (size_t)b * 6 * SPATIAL + soff;

        // natural loads: lane l holds x[c][pixel base+l]  (for softmax-weighted sum)
        float xn[6];
        #pragma unroll
        for (int c = 0; c < 6; ++c) xn[c] = xb[c * SPATIAL + lane];

        // ---- conv1 half 0 (pixels base..base+15): B loaded directly in WMMA layout
        v2f B;
        B.x = xb[c_b0 * SPATIAL + m];                 // rows K0 / K2
        B.y = xb[c_b1 * SPATIAL + m];                 // rows K1 / K3
        v8f h0 = wmma4(A1a, B, C1);
        B.x = xb[4 * SPATIAL + m];                    // row K4 (hi lanes: K6 -> A=0)
        B.y = xb[5 * SPATIAL + m];                    // row K5 (hi lanes: K7 -> A=0)
        h0 = wmma4(A1b, B, h0);

        // ---- conv1 half 1 (pixels base+16..base+31)
        B.x = xb[c_b0 * SPATIAL + 16 + m];
        B.y = xb[c_b1 * SPATIAL + 16 + m];
        v8f h1 = wmma4(A1a, B, C1);
        B.x = xb[4 * SPATIAL + 16 + m];
        B.y = xb[5 * SPATIAL + 16 + m];
        h1 = wmma4(A1b, B, h1);

        // ---- ReLU + conv2, half 0: D rows 0..2 (lanes 0-15) -> B rows K0..K2
        float a0 = fmaxf(h0[0], 0.f);
        float a1 = fmaxf(h0[1], 0.f);
        float a2 = fmaxf(h0[2], 0.f);
        float t  = __shfl_xor(a2, 16, 32);            // row2 -> hi lanes (K2 slot)
        B.x = hi ? t : a0;
        B.y = a1;                                     // hi lanes = K3 row -> A=0
        v8f s0 = wmma4(A2, B, C2);

        // ---- ReLU + conv2, half 1
        a0 = fmaxf(h1[0], 0.f);
        a1 = fmaxf(h1[1], 0.f);
        a2 = fmaxf(h1[2], 0.f);
        t  = __shfl_xor(a2, 16, 32);
        B.x = hi ? t : a0;
        B.y = a1;
        v8f s1 = wmma4(A2, B, C2);

        // ---- merge halves: lane l (l<16) <- s0 row k, lane 16+n <- s1 row k of lane n
        float sc[6];
        #pragma unroll
        for (int k = 0; k < 6; ++k) {
            float th = __shfl_xor(s1[k], 16, 32);
            sc[k] = hi ? th : s0[k];
        }

        // ---- channel softmax fused with weighted sum over channels
        float mx = sc[0];
        #pragma unroll
        for (int k = 1; k < 6; ++k) mx = fmaxf(mx, sc[k]);
        float se = 0.f, xe = 0.f;
        #pragma unroll
        for (int k = 0; k < 6; ++k) {
            float e = __expf(sc[k] - mx);
            se += e;
            xe = fmaf(xn[k], e, xe);
        }
        out[base + lane] = __fdividef(xe, se);
    }
}

extern "C" void kernel_launch(void* const* d_in, const int* in_sizes, int n_in,
                              void* d_out, int out_size, void* d_ws, size_t ws_size,
                              hipStream_t stream) {
    const float* x   = (const float*)d_in[0];
    const float* w1  = (const float*)d_in[1];
    const float* b1  = (const float*)d_in[2];
    const float* g1  = (const float*)d_in[3];
    const float* be1 = (const float*)d_in[4];
    const float* m1  = (const float*)d_in[5];
    const float* v1  = (const float*)d_in[6];
    const float* w2  = (const float*)d_in[7];
    const float* b2  = (const float*)d_in[8];
    const float* g2  = (const float*)d_in[9];
    const float* be2 = (const float*)d_in[10];
    const float* m2  = (const float*)d_in[11];
    const float* v2  = (const float*)d_in[12];
    float* out = (float*)d_out;

    // 2048 blocks x 256 threads = 16384 waves; 262144 blocks / 16384 = 16 iters/wave
    aff_softmax_kernel<<<dim3(2048), dim3(256), 0, stream>>>(
        x, w1, b1, g1, be1, m1, v1, w2, b2, g2, be2, m2, v2, out);
}